// Htne_16509854285882
// MI455X (gfx1250) — compile-verified
//
#include <hip/hip_runtime.h>
#include <cmath>

// HTNE loss: B=4096, H=50, N=10, V=100000, E=128
#define H_   50
#define N_   10
#define E_   128
#define HP   64      // H padded to 4 x 16 WMMA M-tiles
#define NP   16      // N padded to one 16-wide WMMA N-tile
#define LSTR 132     // LDS row stride (floats), padded vs 128 to dodge bank conflicts
#define EPSL 1e-6f

typedef __attribute__((ext_vector_type(2))) float v2f;
typedef __attribute__((ext_vector_type(8))) float v8f;

__device__ __forceinline__ float wave_sum(float v) {
#pragma unroll
  for (int off = 16; off > 0; off >>= 1) v += __shfl_xor(v, off, 32);
  return v;
}
__device__ __forceinline__ float wave_max(float v) {
#pragma unroll
  for (int off = 16; off > 0; off >>= 1) v = fmaxf(v, __shfl_xor(v, off, 32));
  return v;
}

__global__ __launch_bounds__(128) void htne_kernel(
    const int* __restrict__ s_nodes, const int* __restrict__ t_nodes,
    const float* __restrict__ t_times, const int* __restrict__ h_nodes,
    const float* __restrict__ h_times, const float* __restrict__ h_mask,
    const int* __restrict__ n_nodes, const float* __restrict__ emb,
    const float* __restrict__ delta_w, float* __restrict__ out)
{
  __shared__ float s_src[E_];
  __shared__ float s_tar[E_];
  __shared__ float s_his[HP * LSTR];
  __shared__ float s_neg[NP * LSTR];
  __shared__ float s_hsq[HP], s_ash[HP], s_pal[HP], s_wgt[HP];
  __shared__ float s_nsq[NP], s_nmu[NP];
  __shared__ float s_part[4][32];
  __shared__ float s_plambda, s_pmu;

  const int b    = blockIdx.x;
  const int t    = threadIdx.x;
  const int lane = t & 31;
  const int wave = t >> 5;

  // ---------- phase 1: gather embedding rows into LDS (float4, coalesced) ----------
  if (wave == 0) {
    const float4* p = reinterpret_cast<const float4*>(emb + (size_t)s_nodes[b] * E_);
    reinterpret_cast<float4*>(s_src)[lane] = p[lane];
  } else if (wave == 1) {
    const float4* p = reinterpret_cast<const float4*>(emb + (size_t)t_nodes[b] * E_);
    reinterpret_cast<float4*>(s_tar)[lane] = p[lane];
  }
  for (int row = wave; row < HP; row += 4) {
    float4 v = {0.f, 0.f, 0.f, 0.f};
    if (row < H_) {
      int node = h_nodes[b * H_ + row];
      v = reinterpret_cast<const float4*>(emb + (size_t)node * E_)[lane];
    }
    reinterpret_cast<float4*>(&s_his[row * LSTR])[lane] = v;
  }
  for (int row = wave; row < NP; row += 4) {
    float4 v = {0.f, 0.f, 0.f, 0.f};
    if (row < N_) {
      int node = n_nodes[b * N_ + row];
      v = reinterpret_cast<const float4*>(emb + (size_t)node * E_)[lane];
    }
    reinterpret_cast<float4*>(&s_neg[row * LSTR])[lane] = v;
  }
  __syncthreads();

  // ---------- phase 2: per-row stats via wave32 shuffle reductions ----------
  for (int h = wave; h < HP; h += 4) {
    float hsq = 0.f, ds = 0.f, dt = 0.f;
#pragma unroll
    for (int i = 0; i < 4; ++i) {
      int e = lane + 32 * i;
      float x  = s_his[h * LSTR + e];
      float d1 = x - s_src[e];
      float d2 = x - s_tar[e];
      hsq += x * x; ds += d1 * d1; dt += d2 * d2;
    }
    hsq = wave_sum(hsq); ds = wave_sum(ds); dt = wave_sum(dt);
    if (lane == 0) { s_hsq[h] = hsq; s_ash[h] = -ds; s_pal[h] = -dt; }
  }
  for (int n = wave; n < NP; n += 4) {
    float nsq = 0.f, ds = 0.f;
#pragma unroll
    for (int i = 0; i < 4; ++i) {
      int e = lane + 32 * i;
      float x  = s_neg[n * LSTR + e];
      float d1 = x - s_src[e];
      nsq += x * x; ds += d1 * d1;
    }
    nsq = wave_sum(nsq); ds = wave_sum(ds);
    if (lane == 0) { s_nsq[n] = nsq; s_nmu[n] = -ds; }
  }
  float pmu_part = 0.f;
  if (wave == 3) {
#pragma unroll
    for (int i = 0; i < 4; ++i) {
      int e = lane + 32 * i;
      float d = s_src[e] - s_tar[e];
      pmu_part += d * d;
    }
  }
  __syncthreads();

  // ---------- phase 3: softmax + Hawkes weights (wave 0), p_mu (wave 3) ----------
  if (wave == 3) {
    float pmu = wave_sum(pmu_part);
    if (lane == 0) s_pmu = -pmu;
  }
  if (wave == 0) {
    int  h0 = lane, h1 = lane + 32;
    bool v0 = h0 < H_, v1 = h1 < H_;
    float a0 = v0 ? s_ash[h0] : -3.0e38f;
    float a1 = v1 ? s_ash[h1] : -3.0e38f;
    float m  = wave_max(fmaxf(a0, a1));
    float e0 = v0 ? expf(a0 - m) : 0.f;
    float e1 = v1 ? expf(a1 - m) : 0.f;
    float inv = 1.f / wave_sum(e0 + e1);
    float tt    = t_times[b];
    float delta = delta_w[s_nodes[b]];
    float w0 = 0.f, w1 = 0.f;
    if (v0) {
      float dtm = fabsf(tt - h_times[b * H_ + h0]);
      w0 = (e0 * inv) * expf(delta * dtm) * h_mask[b * H_ + h0];
    }
    if (v1) {
      float dtm = fabsf(tt - h_times[b * H_ + h1]);
      w1 = (e1 * inv) * expf(delta * dtm) * h_mask[b * H_ + h1];
    }
    s_wgt[h0] = w0; s_wgt[h1] = w1;
    float pl = w0 * (v0 ? s_pal[h0] : 0.f) + w1 * (v1 ? s_pal[h1] : 0.f);
    pl = wave_sum(pl);
    if (lane == 0) s_plambda = pl;
  }
  __syncthreads();

  // ---------- phase 4: hn = his(64x128) * neg^T(128x16) via V_WMMA_F32_16X16X4_F32 ----
  // wave w owns M-tile rows [16w, 16w+16). Branch-free: EXEC stays all-ones.
  // A frag (16x4 f32): lane<16 -> K=0,1 ; lane>=16 -> K=2,3 of row M=lane%16.
  // B frag (4x16 f32): same K split, N=lane%16.
  v8f c = {0.f, 0.f, 0.f, 0.f, 0.f, 0.f, 0.f, 0.f};
  const int mrow = wave * 16 + (lane & 15);
  const int ncol = lane & 15;
  const int koff = (lane < 16) ? 0 : 2;
  const float* arow = &s_his[mrow * LSTR + koff];
  const float* brow = &s_neg[ncol * LSTR + koff];
#pragma unroll 8
  for (int k = 0; k < E_; k += 4) {
    v2f a  = *reinterpret_cast<const v2f*>(arow + k);
    v2f bb = *reinterpret_cast<const v2f*>(brow + k);
    c = __builtin_amdgcn_wmma_f32_16x16x4_f32(false, a, false, bb,
                                              (short)0, c, false, false);
  }

  // D layout: VGPR r, lane<16 -> M = 16w + r ; lane>=16 -> M = 16w + 8 + r ; N = lane%16
  float part = 0.f;
  const int rbase = wave * 16 + ((lane < 16) ? 0 : 8);
  const float nsqc = s_nsq[ncol];
#pragma unroll
  for (int r = 0; r < 8; ++r) {
    int h = rbase + r;
    float nal = -(s_hsq[h] + nsqc - 2.f * c[r]);   // n_alpha[h][n]
    part += s_wgt[h] * nal;                        // w_h * n_alpha (w=0 for padded h)
  }
  s_part[wave][lane] = part;
  __syncthreads();

  // ---------- phase 5: deterministic final reduction + losses ----------
  if (t == 0) {
    float negloss = 0.f;
    for (int n = 0; n < N_; ++n) {
      float acc = 0.f;
#pragma unroll
      for (int w = 0; w < 4; ++w) acc += s_part[w][n] + s_part[w][n + 16];
      float nl = s_nmu[n] + acc;                   // n_lambda[n]
      float sg = 1.f / (1.f + expf(nl));           // sigmoid(-n_lambda)
      negloss += logf(sg + EPSL);
    }
    float plam = s_pmu + s_plambda;                // p_lambda
    float sgp  = 1.f / (1.f + expf(-plam));        // sigmoid(p_lambda)
    float pos  = -logf(sgp + EPSL);
    out[b] = pos - negloss;
  }
}

extern "C" void kernel_launch(void* const* d_in, const int* in_sizes, int n_in,
                              void* d_out, int out_size, void* d_ws, size_t ws_size,
                              hipStream_t stream) {
  const int*   s_nodes = (const int*)d_in[0];
  const int*   t_nodes = (const int*)d_in[1];
  const float* t_times = (const float*)d_in[2];
  const int*   h_nodes = (const int*)d_in[3];
  const float* h_times = (const float*)d_in[4];
  const float* h_mask  = (const float*)d_in[5];
  const int*   n_nodes = (const int*)d_in[6];
  const float* emb     = (const float*)d_in[7];
  const float* delta_w = (const float*)d_in[8];
  float* out = (float*)d_out;
  const int B = in_sizes[0];   // s_nodes is [B,1] -> B elements

  htne_kernel<<<dim3(B), dim3(128), 0, stream>>>(
      s_nodes, t_nodes, t_times, h_nodes, h_times, h_mask, n_nodes,
      emb, delta_w, out);
}